// RankNCELoss_56178172232252
// MI455X (gfx1250) — compile-verified
//
#include <hip/hip_runtime.h>
#include <math.h>

// ---------------------------------------------------------------------------
// RankNCELoss for N=8192, D=256 fp32.
// Phase 1: sim = Q @ K^T via V_WMMA_F32_16X16X4_F32 (full fp32 precision).
// Phase 2: per-row radix-select of rank-819 / rank-4094 order statistics +
//          stabilized logsumexp. Sorting in the reference is observationally
//          equivalent to rank-range selection, so no sort is performed.
// ---------------------------------------------------------------------------

typedef __attribute__((ext_vector_type(2))) float v2f;
typedef __attribute__((ext_vector_type(8))) float v8f;

#define BM 128
#define BN 64
#define BK 32
#define LDP 33   // padded LDS row stride (odd -> conflict-free b64-ish reads)

__global__ __launch_bounds__(256)
void ranknce_gemm_f32wmma(const float* __restrict__ Q,
                          const float* __restrict__ K,
                          float* __restrict__ S,
                          int N, int D) {
    __shared__ float As[BM * LDP];   // As[r][k], r in [0,BM), k in [0,BK)
    __shared__ float Bs[BN * LDP];   // Bs[n][k]

    const int tid  = threadIdx.x;
    const int lane = tid & 31;
    const int wave = tid >> 5;
    const int rowBase = blockIdx.y * BM;
    const int colBase = blockIdx.x * BN;

    v8f acc[4];
#pragma unroll
    for (int t = 0; t < 4; ++t) acc[t] = {};

    for (int kk = 0; kk < D; kk += BK) {
        // stage A: BM x BK = 4096 floats = 1024 float4, 4 per thread, coalesced
#pragma unroll
        for (int v = 0; v < 4; ++v) {
            int f  = tid + v * 256;
            int r  = f >> 3;           // BK/4 = 8 float4 per row
            int c4 = f & 7;
            float4 val = *(const float4*)(Q + (size_t)(rowBase + r) * D + kk + c4 * 4);
            float* dst = &As[r * LDP + c4 * 4];
            dst[0] = val.x; dst[1] = val.y; dst[2] = val.z; dst[3] = val.w;
        }
        // stage B: BN x BK = 2048 floats = 512 float4, 2 per thread
#pragma unroll
        for (int v = 0; v < 2; ++v) {
            int f  = tid + v * 256;
            int n  = f >> 3;
            int c4 = f & 7;
            float4 val = *(const float4*)(K + (size_t)(colBase + n) * D + kk + c4 * 4);
            float* dst = &Bs[n * LDP + c4 * 4];
            dst[0] = val.x; dst[1] = val.y; dst[2] = val.z; dst[3] = val.w;
        }
        __syncthreads();

        // ISA A layout (16x4 f32): lane<16 holds K={0,1}, lane>=16 holds K={2,3},
        // M = lane&15.  B (4x16): same K split, N = lane&15.
        const int r   = wave * 16 + (lane & 15);
        const int chi = (lane >> 4) * 2;
#pragma unroll
        for (int k0 = 0; k0 < BK; k0 += 4) {
            v2f a;
            a.x = As[r * LDP + k0 + chi + 0];
            a.y = As[r * LDP + k0 + chi + 1];
#pragma unroll
            for (int t = 0; t < 4; ++t) {
                const int n = t * 16 + (lane & 15);
                v2f b;
                b.x = Bs[n * LDP + k0 + chi + 0];
                b.y = Bs[n * LDP + k0 + chi + 1];
                acc[t] = __builtin_amdgcn_wmma_f32_16x16x4_f32(
                    false, a, false, b, (short)0, acc[t], false, false);
            }
        }
        __syncthreads();
    }

    // C/D layout: lanes 0-15 -> M = v, lanes 16-31 -> M = v+8; N = lane&15
    const int outRow  = rowBase + wave * 16 + ((lane >> 4) * 8);
    const int outCol0 = colBase + (lane & 15);
#pragma unroll
    for (int t = 0; t < 4; ++t)
#pragma unroll
        for (int v = 0; v < 8; ++v)
            S[(size_t)(outRow + v) * N + outCol0 + t * 16] = acc[t][v];
}

// ---------------------------------------------------------------------------

__device__ __forceinline__ unsigned fkey(float f) {
    unsigned u = __float_as_uint(f);
    return (u & 0x80000000u) ? ~u : (u | 0x80000000u);  // order-preserving
}
__device__ __forceinline__ float keyToFloat(unsigned key) {
    unsigned u = (key & 0x80000000u) ? (key ^ 0x80000000u) : ~key;
    return __uint_as_float(u);
}

#define MAXN 8192

__global__ __launch_bounds__(256)
void ranknce_loss(const float* __restrict__ S, float* __restrict__ out,
                  int N, int kBottom, int kTop, float invT) {
    __shared__ float    rowv[MAXN];   // 32 KB row cache (LDS-resident passes)
    __shared__ unsigned hist[256];
    __shared__ unsigned sel[2];
    __shared__ float    red[256];

    const int row = blockIdx.x;
    const int tid = threadIdx.x;
    const float* src = S + (size_t)row * N;

    for (int j = tid; j < N; j += 256) rowv[j] = src[j];
    __syncthreads();
    const float lpos = rowv[row];     // diagonal == l_pos
    __syncthreads();
    if (tid == 0) rowv[row] = -INFINITY;  // mask positive; its key sorts last
    __syncthreads();

    // radix select: key of the element at descending rank `k` (0-indexed)
    unsigned keys[2];
    const int ranks[2] = { kBottom, kTop - 1 };   // largest kept, smallest kept
#pragma unroll 1
    for (int q = 0; q < 2; ++q) {
        unsigned prefix = 0;
        int r = ranks[q];
#pragma unroll 1
        for (int shift = 24; shift >= 0; shift -= 8) {
            hist[tid] = 0;
            __syncthreads();
            for (int j = tid; j < N; j += 256) {
                unsigned key = fkey(rowv[j]);
                bool match = (shift == 24) || ((key >> (shift + 8)) == prefix);
                if (match) atomicAdd(&hist[(key >> shift) & 0xFFu], 1u);
            }
            __syncthreads();
            if (tid == 0) {
                unsigned c = 0, rr = (unsigned)r, bin = 0;
                for (int b = 255; b >= 0; --b) {
                    unsigned nc = c + hist[b];
                    if (nc > rr) { bin = (unsigned)b; rr -= c; break; }
                    c = nc;
                }
                sel[0] = bin; sel[1] = rr;
            }
            __syncthreads();
            prefix = (prefix << 8) | sel[0];
            r = (int)sel[1];
            __syncthreads();
        }
        keys[q] = prefix;
    }
    const unsigned keyA = keys[0];   // value at rank kBottom (largest kept)
    const unsigned keyB = keys[1];   // value at rank kTop-1 (smallest kept)

    // stabilized logsumexp; max logit = max(valA, lpos)/T (-10/T always smaller)
    const float m = fmaxf(keyToFloat(keyA), lpos) * invT;

    float partial = 0.0f;
    for (int j = tid; j < N; j += 256) {
        float v = rowv[j];
        unsigned key = fkey(v);            // -inf key < every finite key
        if (key >= keyB && key <= keyA) partial += expf(v * invT - m);
    }
    red[tid] = partial;
    __syncthreads();
    for (int s = 128; s > 0; s >>= 1) {
        if (tid < s) red[tid] += red[tid + s];
        __syncthreads();
    }
    if (tid == 0) {
        const int nFill = N - (kTop - kBottom);      // entries pinned to -10.0
        float St = red[0]
                 + expf(lpos * invT - m)
                 + (float)nFill * expf(-10.0f * invT - m);
        out[row] = m + logf(St) - lpos * invT;
    }
}

// ---------------------------------------------------------------------------

extern "C" void kernel_launch(void* const* d_in, const int* in_sizes, int n_in,
                              void* d_out, int out_size, void* d_ws, size_t ws_size,
                              hipStream_t stream) {
    const float* q = (const float*)d_in[0];
    const float* k = (const float*)d_in[1];
    float* out = (float*)d_out;

    const int N = out_size;               // 8192
    const int D = in_sizes[0] / N;        // 256

    const int numNeg  = N - 1;
    int kTop    = (int)(numNeg * 0.5f);   // 4095 (matches python int())
    if (kTop < 1) kTop = 1;
    int kBottom = (int)(numNeg * 0.1f);   // 819
    if (kBottom < 0) kBottom = 0;

    float* sim = (float*)d_ws;            // needs N*N*4 = 256 MB scratch

    dim3 g1(N / BN, N / BM);
    ranknce_gemm_f32wmma<<<g1, 256, 0, stream>>>(q, k, sim, N, D);

    ranknce_loss<<<N, 256, 0, stream>>>(sim, out, N, kBottom, kTop,
                                        1.0f / 0.07f);
}